// GAT_75995151335797
// MI455X (gfx1250) — compile-verified
//
#include <hip/hip_runtime.h>
#include <math.h>

typedef __attribute__((ext_vector_type(2))) float v2f;
typedef __attribute__((ext_vector_type(8))) float v8f;

#define NEG_SLOPE 0.2f

// float atomic max via signed-max / unsigned-min bit trick (init = -inf bits)
__device__ __forceinline__ void atomicMaxFloat(float* addr, float val) {
  if (val >= 0.0f) atomicMax((int*)addr, __float_as_int(val));
  else             atomicMin((unsigned int*)addr, __float_as_uint(val));
}

// fetch (src,dst): from packed int32 pairs if available, else from int64 list
__device__ __forceinline__ int2 get_edge(const int2* __restrict__ e32,
                                         const long long* __restrict__ ei,
                                         int e, int E) {
  if (e32) return e32[e];
  int2 r;
  if (e < E) { r.x = (int)ei[e]; r.y = (int)ei[E + e]; }
  else       { r.x = r.y = e - E; }            // self loops appended
  return r;
}

__global__ void k_fill(float* p, int n, float v) {
  int i = blockIdx.x * blockDim.x + threadIdx.x;
  if (i < n) p[i] = v;
}

// pack int64 [2,E] edge list (+ self loops) into int2 pairs
__global__ void k_prep(const long long* __restrict__ ei, int2* __restrict__ e32,
                       int E, int Et) {
  int e = blockIdx.x * blockDim.x + threadIdx.x;
  if (e >= Et) return;
  int2 r;
  if (e < E) { r.x = (int)ei[e]; r.y = (int)ei[E + e]; }
  else       { r.x = r.y = e - E; }
  e32[e] = r;
}

// h1[Nn x 64] = x[Nn x 3] @ W1[3 x 64]  via V_WMMA_F32_16X16X4_F32 (K padded 3->4)
// A 16x4 f32: lanes 0-15 hold K={0,1} in vgpr{0,1}; lanes 16-31 hold K={2,3}
// B 4x16 f32: vgpr0 = rows {K0 | K2}, vgpr1 = rows {K1 | K3}
// D 16x16 f32: vgpr r -> M = r (lanes 0-15) / M = r+8 (lanes 16-31), N = lane%16
__global__ void k_h1_wmma(const float* __restrict__ x, const float* __restrict__ W1,
                          float* __restrict__ h1, int Nn) {
  const int lane = threadIdx.x;          // one wave32 per block
  const int m0   = blockIdx.x * 16;
  const int lo   = lane & 15;
  const bool hi  = lane >= 16;
  const bool full = (m0 + 16) <= Nn;     // wave-uniform full-tile fast path
  int row = m0 + lo;
  float x0 = 0.f, x1 = 0.f, x2 = 0.f;
  if (row < Nn) { x0 = x[row*3+0]; x1 = x[row*3+1]; x2 = x[row*3+2]; }
  v2f a; a.x = hi ? x2 : x0; a.y = hi ? 0.f : x1;   // K=3 zero-padded
#pragma unroll
  for (int nt = 0; nt < 4; ++nt) {
    int n = nt * 16 + lo;
    v2f b; b.x = hi ? W1[2*64+n] : W1[0*64+n];
           b.y = hi ? 0.f        : W1[1*64+n];
    v8f c = {};
    c = __builtin_amdgcn_wmma_f32_16x16x4_f32(false, a, false, b,
                                              (short)0, c, false, false);
    union { v8f v; float f[8]; } u; u.v = c;
    int mb = m0 + (hi ? 8 : 0);
    if (full) {
#pragma unroll
      for (int r = 0; r < 8; ++r) h1[(mb + r)*64 + n] = u.f[r];
    } else {
#pragma unroll
      for (int r = 0; r < 8; ++r) if (mb + r < Nn) h1[(mb + r)*64 + n] = u.f[r];
    }
  }
}

// asrc1/adst1[n][h] = sum_c h1[n][h*8+c] * att_{src,dst}1[h][c]
__global__ void k_att1(const float* __restrict__ h1, const float* __restrict__ as,
                       const float* __restrict__ ad, float* __restrict__ asrc,
                       float* __restrict__ adst, int Nn) {
  int t = blockIdx.x * blockDim.x + threadIdx.x;
  if (t >= Nn * 8) return;
  int n = t >> 3, h = t & 7;
  float s = 0.f, d = 0.f;
#pragma unroll
  for (int c = 0; c < 8; ++c) {
    float v = h1[n*64 + h*8 + c];
    s += v * as[h*8 + c];
    d += v * ad[h*8 + c];
  }
  asrc[t] = s; adst[t] = d;
}

// edge pass 1 (conv1): per (edge, head) atomic max of leaky-relu logits
__global__ void k_max1(const int2* __restrict__ e32, const long long* __restrict__ ei,
                       const float* __restrict__ asrc, const float* __restrict__ adst,
                       float* __restrict__ emax, int E, int Et) {
  int idx = blockIdx.x * blockDim.x + threadIdx.x;
  if (idx >= Et * 8) return;
  int e = idx >> 3, h = idx & 7;
  int2 sd = get_edge(e32, ei, e, E);
  float v = asrc[sd.x*8 + h] + adst[sd.y*8 + h];
  v = v > 0.f ? v : NEG_SLOPE * v;
  atomicMaxFloat(&emax[sd.y*8 + h], v);
}

// edge pass 2 (conv1): one thread per (edge, channel-pair) -> one wave32 == one
// edge. Edge loads are wave-uniform (attention/emax vary by head); the 64 atomics per edge hit
// 64 consecutive floats of out1[dst] -> fully coalesced L2 atomic traffic.
__global__ void k_agg1(const int2* __restrict__ e32, const long long* __restrict__ ei,
                       const float* __restrict__ asrc, const float* __restrict__ adst,
                       const float* __restrict__ emax, const float* __restrict__ h1,
                       float* __restrict__ denom, float* __restrict__ out1,
                       int E, int Et) {
  int idx = blockIdx.x * blockDim.x + threadIdx.x;
  if (idx >= Et * 32) return;
  int e = idx >> 5, l = idx & 31;      // l = lane, j = 2*l, h = l>>2
  int j = l << 1, h = l >> 2;
  if (l == 0 && e32 && e + 4096 < Et) __builtin_prefetch(&e32[e + 4096], 0, 0);
  int2 sd = get_edge(e32, ei, e, E);
  float v = asrc[sd.x*8 + h] + adst[sd.y*8 + h];
  v = v > 0.f ? v : NEG_SLOPE * v;
  float ee = __expf(v - emax[sd.y*8 + h]);
  if ((l & 3) == 0) atomicAdd(&denom[sd.y*8 + h], ee);
  const float2 hv = *reinterpret_cast<const float2*>(&h1[sd.x*64 + j]);
  atomicAdd(&out1[sd.y*64 + j],     ee * hv.x);
  atomicAdd(&out1[sd.y*64 + j + 1], ee * hv.y);
}

// normalize + bias + ELU -> input of conv2
__global__ void k_fin1(const float* __restrict__ out1, const float* __restrict__ denom,
                       const float* __restrict__ b1, float* __restrict__ h2in, int Nn) {
  int idx = blockIdx.x * blockDim.x + threadIdx.x;
  if (idx >= Nn * 64) return;
  int n = idx >> 6, h = (idx >> 3) & 7;
  float v = out1[idx] / denom[n*8 + h] + b1[idx & 63];
  h2in[idx] = v > 0.f ? v : expm1f(v);
}

// h2 = h2in @ W2 (64->4), plus attention logits (1 head)
__global__ void k_h2(const float* __restrict__ h2in, const float* __restrict__ W2,
                     const float* __restrict__ as2, const float* __restrict__ ad2,
                     float* __restrict__ h2, float* __restrict__ asrc2,
                     float* __restrict__ adst2, int Nn) {
  int n = blockIdx.x * blockDim.x + threadIdx.x;
  if (n >= Nn) return;
  float acc[4] = {0.f, 0.f, 0.f, 0.f};
  for (int k = 0; k < 64; ++k) {
    float v = h2in[n*64 + k];
#pragma unroll
    for (int j = 0; j < 4; ++j) acc[j] += v * W2[k*4 + j];
  }
  float s = 0.f, d = 0.f;
#pragma unroll
  for (int j = 0; j < 4; ++j) {
    h2[n*4 + j] = acc[j];
    s += acc[j] * as2[j];
    d += acc[j] * ad2[j];
  }
  asrc2[n] = s; adst2[n] = d;
}

__global__ void k_max2(const int2* __restrict__ e32, const long long* __restrict__ ei,
                       const float* __restrict__ asrc2, const float* __restrict__ adst2,
                       float* __restrict__ emax2, int E, int Et) {
  int e = blockIdx.x * blockDim.x + threadIdx.x;
  if (e >= Et) return;
  int2 sd = get_edge(e32, ei, e, E);
  float v = asrc2[sd.x] + adst2[sd.y];
  v = v > 0.f ? v : NEG_SLOPE * v;
  atomicMaxFloat(&emax2[sd.y], v);
}

// edge pass 2 (conv2): one thread per edge, float4 gather of h2[src]
__global__ void k_agg2(const int2* __restrict__ e32, const long long* __restrict__ ei,
                       const float* __restrict__ asrc2, const float* __restrict__ adst2,
                       const float* __restrict__ emax2, const float* __restrict__ h2,
                       float* __restrict__ denom2, float* __restrict__ out2,
                       int E, int Et) {
  int e = blockIdx.x * blockDim.x + threadIdx.x;
  if (e >= Et) return;
  int2 sd = get_edge(e32, ei, e, E);
  float v = asrc2[sd.x] + adst2[sd.y];
  v = v > 0.f ? v : NEG_SLOPE * v;
  float ee = __expf(v - emax2[sd.y]);
  atomicAdd(&denom2[sd.y], ee);
  const float4 hv = *reinterpret_cast<const float4*>(&h2[sd.x*4]);
  atomicAdd(&out2[sd.y*4 + 0], ee * hv.x);
  atomicAdd(&out2[sd.y*4 + 1], ee * hv.y);
  atomicAdd(&out2[sd.y*4 + 2], ee * hv.z);
  atomicAdd(&out2[sd.y*4 + 3], ee * hv.w);
}

__global__ void k_fin2(const float* __restrict__ out2, const float* __restrict__ denom2,
                       const float* __restrict__ b2, float* __restrict__ feat, int Nn) {
  int idx = blockIdx.x * blockDim.x + threadIdx.x;
  if (idx >= Nn * 4) return;
  int n = idx >> 2;
  float v = out2[idx] / denom2[n] + b2[idx & 3];
  feat[idx] = v > 0.f ? v : expm1f(v);
}

// logits = flatten(feat) @ fc_W + fc_b ; log_softmax -> out[4]
__global__ void k_logits(const float* __restrict__ feat, const float* __restrict__ fcW,
                         const float* __restrict__ fcb, float* __restrict__ out, int n) {
  __shared__ float red[256][4];
  int tid = threadIdx.x;
  float acc[4] = {0.f, 0.f, 0.f, 0.f};
  for (int i = tid; i < n; i += 256) {
    float f = feat[i];
#pragma unroll
    for (int j = 0; j < 4; ++j) acc[j] += f * fcW[i*4 + j];
  }
#pragma unroll
  for (int j = 0; j < 4; ++j) red[tid][j] = acc[j];
  __syncthreads();
  for (int s = 128; s > 0; s >>= 1) {
    if (tid < s) {
#pragma unroll
      for (int j = 0; j < 4; ++j) red[tid][j] += red[tid + s][j];
    }
    __syncthreads();
  }
  if (tid == 0) {
    float lg[4]; float m = -__builtin_inff();
#pragma unroll
    for (int j = 0; j < 4; ++j) { lg[j] = red[0][j] + fcb[j]; m = fmaxf(m, lg[j]); }
    float se = 0.f;
#pragma unroll
    for (int j = 0; j < 4; ++j) se += __expf(lg[j] - m);
    float lse = m + logf(se);
#pragma unroll
    for (int j = 0; j < 4; ++j) out[j] = lg[j] - lse;
  }
}

static inline int cdiv(long long a, long long b) { return (int)((a + b - 1) / b); }

extern "C" void kernel_launch(void* const* d_in, const int* in_sizes, int n_in,
                              void* d_out, int out_size, void* d_ws, size_t ws_size,
                              hipStream_t stream) {
  (void)n_in; (void)out_size;
  const float*     x   = (const float*)d_in[0];
  const long long* ei  = (const long long*)d_in[1];   // int64 [2,E] flat
  const float*     W1  = (const float*)d_in[2];
  const float*     as1 = (const float*)d_in[3];
  const float*     ad1 = (const float*)d_in[4];
  const float*     b1  = (const float*)d_in[5];
  const float*     W2  = (const float*)d_in[6];
  const float*     as2 = (const float*)d_in[7];
  const float*     ad2 = (const float*)d_in[8];
  const float*     b2  = (const float*)d_in[9];
  const float*     fcW = (const float*)d_in[10];
  const float*     fcb = (const float*)d_in[11];
  float* out = (float*)d_out;

  const int Nn = in_sizes[0] / 3;     // 2500
  const int E  = in_sizes[1] / 2;     // 1,500,000
  const int Et = E + Nn;              // + self loops

  // workspace layout (floats)
  float* ws     = (float*)d_ws;
  float* h1     = ws;                 // Nn*64
  float* asrc1  = h1    + Nn*64;      // Nn*8
  float* adst1  = asrc1 + Nn*8;       // Nn*8
  float* emax1  = adst1 + Nn*8;       // Nn*8   (init -inf)
  float* denom1 = emax1 + Nn*8;       // Nn*8   (init 0)
  float* out1   = denom1+ Nn*8;       // Nn*64  (init 0, contiguous w/ denom1)
  float* h2in   = out1  + Nn*64;      // Nn*64
  float* h2     = h2in  + Nn*64;      // Nn*4
  float* asrc2  = h2    + Nn*4;       // Nn
  float* adst2  = asrc2 + Nn;         // Nn
  float* emax2  = adst2 + Nn;         // Nn     (init -inf)
  float* denom2 = emax2 + Nn;         // Nn     (init 0)
  float* out2   = denom2+ Nn;         // Nn*4   (init 0, contiguous w/ denom2)
  float* feat   = out2  + Nn*4;       // Nn*4
  float* wsend  = feat  + Nn*4;       // = ws + Nn*240

  // optional packed int32 edge pairs (12 MB) if workspace allows
  const size_t base_bytes = (size_t)(wsend - ws) * sizeof(float);
  int2* e32 = nullptr;
  if (ws_size >= base_bytes + (size_t)Et * sizeof(int2))
    e32 = reinterpret_cast<int2*>(wsend);

  const int TB = 256;
  const float NEGINF = -__builtin_inff();

  // init accumulators (deterministic each call)
  k_fill<<<cdiv(Nn*8, TB), TB, 0, stream>>>(emax1, Nn*8, NEGINF);
  k_fill<<<cdiv(Nn, TB), TB, 0, stream>>>(emax2, Nn, NEGINF);
  k_fill<<<cdiv((long long)Nn*(8+64), TB), TB, 0, stream>>>(denom1, Nn*(8+64), 0.f);
  k_fill<<<cdiv((long long)Nn*(1+4), TB), TB, 0, stream>>>(denom2, Nn*(1+4), 0.f);
  if (e32) k_prep<<<cdiv(Et, TB), TB, 0, stream>>>(ei, e32, E, Et);

  // conv1
  k_h1_wmma<<<cdiv(Nn, 16), 32, 0, stream>>>(x, W1, h1, Nn);
  k_att1<<<cdiv((long long)Nn*8, TB), TB, 0, stream>>>(h1, as1, ad1, asrc1, adst1, Nn);
  k_max1<<<cdiv((long long)Et*8, TB), TB, 0, stream>>>(e32, ei, asrc1, adst1, emax1, E, Et);
  k_agg1<<<cdiv((long long)Et*32, TB), TB, 0, stream>>>(e32, ei, asrc1, adst1, emax1, h1,
                                                        denom1, out1, E, Et);
  k_fin1<<<cdiv((long long)Nn*64, TB), TB, 0, stream>>>(out1, denom1, b1, h2in, Nn);

  // conv2
  k_h2<<<cdiv(Nn, TB), TB, 0, stream>>>(h2in, W2, as2, ad2, h2, asrc2, adst2, Nn);
  k_max2<<<cdiv(Et, TB), TB, 0, stream>>>(e32, ei, asrc2, adst2, emax2, E, Et);
  k_agg2<<<cdiv(Et, TB), TB, 0, stream>>>(e32, ei, asrc2, adst2, emax2, h2,
                                          denom2, out2, E, Et);
  k_fin2<<<cdiv((long long)Nn*4, TB), TB, 0, stream>>>(out2, denom2, b2, feat, Nn);

  // fc + log_softmax
  k_logits<<<1, 256, 0, stream>>>(feat, fcW, fcb, out, Nn*4);
}